// NNFreeverbModule_28226525070336
// MI455X (gfx1250) — compile-verified
//
#include <hip/hip_runtime.h>

typedef float v2f __attribute__((ext_vector_type(2)));
typedef float v8f __attribute__((ext_vector_type(8)));

// ---------------------------------------------------------------------------
// Phase 1: comb bank. Comb i with lag D: b[t] = x[t] + fb*b[t-D], out[t]=b[t-D].
// Decomposed into D independent residue chains; lane L takes residue r0+L so
// every chain step issues fully coalesced loads/stores (t = r0+L+k*D).
// ---------------------------------------------------------------------------
__global__ void freeverb_comb_kernel(const float* __restrict__ x,
                                     const float* __restrict__ fbL,
                                     const float* __restrict__ fbR,
                                     const int*   __restrict__ dL,
                                     const int*   __restrict__ dR,
                                     float* __restrict__ c, int T)
{
    const int comb = blockIdx.x;                         // 0..15 (8 L, 8 R)
    const int r    = blockIdx.y * blockDim.x + threadIdx.x;
    const int   D  = (comb < 8) ? dL[comb] : dR[comb - 8];
    const float fb = (comb < 8) ? fbL[comb] : fbR[comb - 8];
    if (r >= D) return;
    float* __restrict__ cc = c + (size_t)comb * (size_t)T;
    float prev = 0.0f;                                   // b[t-D] for t<D is 0
    for (int t = r; t < T; t += D) {
        __builtin_prefetch(&x[t + 4 * D], 0, 0);         // global_prefetch_b8
        cc[t] = prev;                                    // comb output at t
        prev  = fmaf(prev, fb, x[t]);                    // b[t] = x[t] + fb*b[t-D]
    }
}

// ---------------------------------------------------------------------------
// Phase 2: y[ch][t] = sum of 8 comb outputs, done on the matrix core.
// V_WMMA_F32_16X16X4_F32, A = ones(16x4):  D[m][n] = sum_k B[k][n].
// B layout (32-bit, 4xK x 16xN): lanes 0-15 hold N=0..15 with K=0 (VGPR0),
// K=1 (VGPR1); lanes 16-31 hold K=2/K=3.  Two accumulated WMMAs cover all
// 8 combs; result row M=0 (VGPR0, lanes 0-15) holds the 16 sums in fp32.
// ---------------------------------------------------------------------------
__global__ void freeverb_sum_kernel(const float* __restrict__ c,
                                    float* __restrict__ y, int T)
{
    const int ch   = blockIdx.y;
    const int lane = threadIdx.x & 31;
    const int wave = threadIdx.x >> 5;
    const int n    = lane & 15;
    const int hi   = lane >> 4;                          // 0: K=0,1  1: K=2,3
    const int t    = blockIdx.x * 128 + wave * 16 + n;
    const int tl   = (t < T) ? t : (T - 1);              // clamp loads; EXEC stays all-1s

    const float* __restrict__ base = c + (size_t)ch * 8u * (size_t)T;
    v2f b0, b1;
    b0.x = base[(size_t)(hi * 2 + 0) * (size_t)T + tl];  // combs 0..3
    b0.y = base[(size_t)(hi * 2 + 1) * (size_t)T + tl];
    b1.x = base[(size_t)(hi * 2 + 4) * (size_t)T + tl];  // combs 4..7
    b1.y = base[(size_t)(hi * 2 + 5) * (size_t)T + tl];

    v2f a = {1.0f, 1.0f};                                // all-ones A matrix
    v8f acc = {};
    acc = __builtin_amdgcn_wmma_f32_16x16x4_f32(false, a, false, b0,
                                                (short)0, acc, false, false);
    acc = __builtin_amdgcn_wmma_f32_16x16x4_f32(false, a, false, b1,
                                                (short)0, acc, false, false);
    if (lane < 16 && t < T)
        y[(size_t)ch * (size_t)T + t] = acc[0];          // row M=0, N=lane
}

// ---------------------------------------------------------------------------
// Phase 3 (x4, ping-pong): one allpass stage for both channels.
// w[t] = y[t] + fb*w[t-D]; out[t] = w[t-D] - y[t].  D residue chains,
// each <= ceil(T/225)=292 dependent FMAs (the critical path of the problem).
// ---------------------------------------------------------------------------
__global__ void freeverb_ap_kernel(const float* __restrict__ yin,
                                   float* __restrict__ yout,
                                   const float* __restrict__ fbLp,
                                   const float* __restrict__ fbRp,
                                   const int*   __restrict__ dLp,
                                   const int*   __restrict__ dRp,
                                   int stage, int T)
{
    const int ch = blockIdx.y;
    const int r  = blockIdx.x * blockDim.x + threadIdx.x;
    const int   D  = ch ? dRp[stage] : dLp[stage];
    const float fb = ch ? fbRp[stage] : fbLp[stage];
    if (r >= D) return;
    const float* __restrict__ in  = yin  + (size_t)ch * (size_t)T;
    float*       __restrict__ out = yout + (size_t)ch * (size_t)T;
    float w = 0.0f;                                      // w[t-D] for t<D is 0
    for (int t = r; t < T; t += D) {
        __builtin_prefetch(&in[t + 4 * D], 0, 0);
        const float yv = in[t];
        out[t] = w - yv;                                 // -y + buf_old
        w = fmaf(w, fb, yv);                             // buf_new = y + fb*buf_old
    }
}

// ---------------------------------------------------------------------------
// Phase 4: stereo wet/dry mix, interleaved [T,2] output.
// ---------------------------------------------------------------------------
__global__ void freeverb_mix_kernel(const float* __restrict__ y,
                                    const float* __restrict__ x,
                                    const float* __restrict__ wet_p,
                                    const float* __restrict__ dry_p,
                                    const float* __restrict__ width_p,
                                    float* __restrict__ out, int T)
{
    const int t = blockIdx.x * blockDim.x + threadIdx.x;
    if (t >= T) return;
    const float wet = wet_p[0], dry = dry_p[0], width = width_p[0];
    const float wet1 = wet * (width + 1.0f) * 0.5f;
    const float wet2 = wet * (1.0f - width) * 0.5f;
    const float yL = y[t], yR = y[(size_t)T + t], xi = x[t];
    out[2 * t + 0] = yL * wet1 + yR * wet2 + xi * dry;
    out[2 * t + 1] = yR * wet1 + yL * wet2 + xi * dry;
}

// ---------------------------------------------------------------------------
// Host launcher: comb -> sum(WMMA) -> 4x allpass (ping-pong) -> mix.
// Workspace: c[16*T] | y0[2*T] | y1[2*T]  (20*T floats = 5.25 MB @ T=65536).
// ---------------------------------------------------------------------------
extern "C" void kernel_launch(void* const* d_in, const int* in_sizes, int n_in,
                              void* d_out, int out_size, void* d_ws, size_t ws_size,
                              hipStream_t stream) {
    const float* x            = (const float*)d_in[0];
    const float* comb_fb_L    = (const float*)d_in[1];
    const float* comb_fb_R    = (const float*)d_in[2];
    const float* ap_fb_L      = (const float*)d_in[3];
    const float* ap_fb_R      = (const float*)d_in[4];
    const float* wet          = (const float*)d_in[5];
    const float* dry          = (const float*)d_in[6];
    const float* width        = (const float*)d_in[7];
    const int*   comb_delay_L = (const int*)d_in[8];
    const int*   comb_delay_R = (const int*)d_in[9];
    const int*   ap_delay_L   = (const int*)d_in[10];
    const int*   ap_delay_R   = (const int*)d_in[11];
    const int T = in_sizes[0];

    float* c  = (float*)d_ws;                 // 16*T
    float* y0 = c  + (size_t)16 * (size_t)T;  // 2*T
    float* y1 = y0 + (size_t)2  * (size_t)T;  // 2*T

    // Comb bank: 16 combs x up-to-4117 residues (hard-coded max lag = 4117).
    dim3 cgrid(16, (4117 + 255) / 256);
    freeverb_comb_kernel<<<cgrid, 256, 0, stream>>>(
        x, comb_fb_L, comb_fb_R, comb_delay_L, comb_delay_R, c, T);

    // WMMA reduction: each wave of 32 produces 16 sums; 128 samples per block.
    dim3 sgrid((T + 127) / 128, 2);
    freeverb_sum_kernel<<<sgrid, 256, 0, stream>>>(c, y0, T);

    // 4 serial allpass stages, L and R concurrently; max lag = 579.
    dim3 agrid((579 + 255) / 256, 2);
    float* pin = y0; float* pout = y1;
    for (int s = 0; s < 4; ++s) {
        freeverb_ap_kernel<<<agrid, 256, 0, stream>>>(
            pin, pout, ap_fb_L, ap_fb_R, ap_delay_L, ap_delay_R, s, T);
        float* tmp = pin; pin = pout; pout = tmp;
    }

    freeverb_mix_kernel<<<(T + 255) / 256, 256, 0, stream>>>(
        pin, x, wet, dry, width, (float*)d_out, T);
}